// CARAFE_72181220376848
// MI455X (gfx1250) — compile-verified
//
#include <hip/hip_runtime.h>
#include <stdint.h>

// CARAFE (K=5, G=1, S=2) on gfx1250 via V_WMMA_F32_16X16X4_F32 + async global->LDS.
// Per low-res pixel: out[c, sub] = sum_i patch[c,i] * ker[i,sub]
//   -> GEMM M=256(ch), N=4(sub, replicated to 16 cols), K=25 (padded to 28).
// Memory-bound (~87 MB @ 23.3 TB/s ~ 3.7 us): async LDS staging (ASYNCcnt),
// register-resident B fragments, coalesced b128 output stores.

typedef float v2f __attribute__((ext_vector_type(2)));
typedef float v8f __attribute__((ext_vector_type(8)));

#define KK    5
#define NCH   256
#define HIN   64
#define WIN   64
#define HOUT  128
#define WOUT  128
#define TILE  4          // low-res pixels per tile side
#define HALO  8          // TILE + KK - 1
#define FSTR  68         // channel stride: 8B-aligned rows for async b64,
                         // (4m+C)%64 banks -> conflict-free A reads
#define CCH   128        // channels per LDS chunk (2 chunks of 128)
#define KPAD  28         // 25 taps padded to 28 (multiple of 4)
#define NPIX  16         // TILE*TILE pixels per workgroup

__device__ __forceinline__ uint32_t lds_off(const void* p) {
    // flat shared pointer: addr[31:0] is the LDS byte offset (ISA aperture map)
    return (uint32_t)(uintptr_t)p;
}

__launch_bounds__(128)
__global__ void carafe_wmma_kernel(const float* __restrict__ feat,
                                   const float* __restrict__ masks,
                                   float* __restrict__ out)
{
    __shared__ float sF[CCH * FSTR];        // 34,816 B feature halo tile (chunked)
    __shared__ float sK[NPIX * KPAD * 4];   //  7,168 B kernel weights [pix][28][4]
    __shared__ float sO[4][16][2][8];       //  4,096 B per-wave output restage

    const int tid  = threadIdx.x;
    const int lane = tid & 31;
    const int wave = tid >> 5;              // wave = h_local row (0..3)

    int blk = blockIdx.x;
    const int wb = blk & 15;
    const int hb = (blk >> 4) & 15;
    const int n  = blk >> 8;

    // ---- masks -> sK[pix][tap][sub]: async global->LDS with per-lane scatter ----
    for (int idx = tid; idx < 25 * 64; idx += 128) {
        int kp = idx >> 6;                  // tap 0..24
        int r  = idx & 63;                  // 8x8 hi-res pixels of this tile
        int oy = r >> 3, ox = r & 7;
        const float* gp =
            masks + (((size_t)n * 25 + kp) * HOUT + hb * 8 + oy) * WOUT + wb * 8 + ox;
        int pix = (oy >> 1) * 4 + (ox >> 1);
        int sb  = (oy & 1) * 2 + (ox & 1);
        uint32_t lofs = lds_off(&sK[pix * (KPAD * 4) + kp * 4 + sb]);
        asm volatile("global_load_async_to_lds_b32 %0, %1, off"
                     :: "v"(lofs), "v"((uint64_t)(uintptr_t)gp) : "memory");
    }
    // zero-pad taps 25..27 (disjoint from async destinations)
    for (int idx = tid; idx < NPIX * 12; idx += 128) {
        int pix = idx / 12;
        int rem = idx - pix * 12;
        sK[pix * (KPAD * 4) + 100 + rem] = 0.0f;
    }
    asm volatile("s_wait_asynccnt 0" ::: "memory");
    __syncthreads();

    // ---- preload B fragments (4 pixels x 7 K-chunks) into registers ----
    // B 4x16 layout: VGPR0 = rows {kb+0 | kb+2}, VGPR1 = rows {kb+1 | kb+3},
    // column n striped across lanes 0..15 (sub replicated: n%4).
    const int sub   = lane & 3;
    const int khalf = (lane >= 16) ? 2 : 0;
    v2f Bf[TILE][7];
#pragma unroll
    for (int wl = 0; wl < TILE; ++wl) {
        const float* kb = &sK[(wave * 4 + wl) * (KPAD * 4)];
#pragma unroll
        for (int kc = 0; kc < 7; ++kc) {
            int k0 = kc * 4 + khalf;
            v2f t;
            t.x = kb[k0 * 4 + sub];
            t.y = kb[(k0 + 1) * 4 + sub];
            Bf[wl][kc] = t;
        }
    }

    // ---- per-lane A geometric offsets: tap i -> (i/5)*HALO + i%5 ----
    int dA0[7], dA1[7];
#pragma unroll
    for (int kc = 0; kc < 7; ++kc) {
        int i0 = kc * 4 + khalf;
        int i1 = i0 + 1;
        dA0[kc] = (i0 < 25) ? ((i0 / 5) * HALO + (i0 % 5)) : 0;
        dA1[kc] = (i1 < 25) ? ((i1 / 5) * HALO + (i1 % 5)) : 0;
    }

    const int gy0 = hb * TILE - 2;
    const int gx0 = wb * TILE - 2;          // even -> float pairs are 8B aligned
    const int m   = lane & 15;              // channel row within 16-ch M-tile

    for (int ch = 0; ch < NCH / CCH; ++ch) {
        // ---- feature halo chunk: async b64 global->LDS; zeros for OOB pairs ----
        for (int idx = tid; idx < CCH * 32; idx += 128) {
            int c = idx >> 5, r = idx & 31;
            int y = r >> 2, xp = r & 3;
            int gy = gy0 + y, gx = gx0 + xp * 2;   // pair never straddles border
            int o  = c * FSTR + y * HALO + xp * 2;
            if (gy >= 0 && gy < HIN && gx >= 0 && gx < WIN) {
                const float* gp =
                    feat + (((size_t)n * NCH + ch * CCH + c) * HIN + gy) * WIN + gx;
                uint32_t lofs = lds_off(&sF[o]);
                asm volatile("global_load_async_to_lds_b64 %0, %1, off"
                             :: "v"(lofs), "v"((uint64_t)(uintptr_t)gp) : "memory");
            } else {
                sF[o]     = 0.0f;
                sF[o + 1] = 0.0f;
            }
        }
        asm volatile("s_wait_asynccnt 0" ::: "memory");
        __syncthreads();

        for (int mt = 0; mt < CCH / 16; ++mt) {
            const float* fc = &sF[(mt * 16 + m) * FSTR];
#pragma unroll
            for (int wl = 0; wl < TILE; ++wl) {
                const int pbase = wave * HALO + wl;     // patch origin in tile
                v8f acc = {0.f, 0.f, 0.f, 0.f, 0.f, 0.f, 0.f, 0.f};
#pragma unroll
                for (int kc = 0; kc < 7; ++kc) {
                    v2f A;
                    if (kc < 6) {
                        A.x = fc[pbase + dA0[kc]];
                        A.y = fc[pbase + dA1[kc]];
                    } else {
                        // taps 24..27: only tap 24 (lanes<16, first elem) is real;
                        // B rows 25..27 are zero anyway.
                        float t = fc[pbase + dA0[6]];
                        A.x = (khalf == 0) ? t : 0.0f;
                        A.y = 0.0f;
                    }
                    acc = __builtin_amdgcn_wmma_f32_16x16x4_f32(
                        false, A, false, Bf[wl][kc], (short)0, acc, false, false);
                }
                // D: lane l, reg r -> D[M = r + (l>=16?8:0), N = l&15].
                // Useful columns N<4 (subs); restage via LDS for coalescing.
                if ((lane & 15) < 4) {
                    const int crow = (lane >= 16) ? 8 : 0;
                    const int a  = sub >> 1;
                    const int bb = sub & 1;
#pragma unroll
                    for (int r = 0; r < 8; ++r)
                        sO[wave][crow + r][a][wl * 2 + bb] = acc[r];
                }
            }
            __syncthreads();

            // ---- coalesced b128 output stores: 64 float4 per wave ----
            const int cbase = ch * CCH + mt * 16;
#pragma unroll
            for (int it = 0; it < 2; ++it) {
                int q  = it * 32 + lane;    // 0..63
                int c  = q >> 2;
                int a  = (q >> 1) & 1;
                int xg = q & 1;
                float4 vv = *(const float4*)&sO[wave][c][a][xg * 4];
                size_t gaddr =
                    (((size_t)(n * NCH + cbase + c)) * HOUT + (hb * 8 + wave * 2 + a)) * WOUT
                    + (wb * 8 + xg * 4);
                *(float4*)&out[gaddr] = vv;
            }
            __syncthreads();
        }
    }
}

extern "C" void kernel_launch(void* const* d_in, const int* in_sizes, int n_in,
                              void* d_out, int out_size, void* d_ws, size_t ws_size,
                              hipStream_t stream)
{
    const float* feat  = (const float*)d_in[0];   // (4,256,64,64) fp32
    const float* masks = (const float*)d_in[1];   // (4,25,128,128) fp32
    float* out = (float*)d_out;                   // (4,256,128,128) fp32

    dim3 grid(4 * 16 * 16);   // n * hb * wb
    dim3 block(128);          // 4 waves (wave32); wave = pixel row of 4x4 tile
    carafe_wmma_kernel<<<grid, block, 0, stream>>>(feat, masks, out);
}